// GraphConvolutionLayer_9105330668111
// MI455X (gfx1250) — compile-verified
//
#include <hip/hip_runtime.h>

// ============================================================================
// GraphConvolutionLayer, MI455X (gfx1250) optimized.
//
// Reference:
//   msg   = concat(x[src], x[dst]) @ W_msg + b_msg          [E=640000, 128]
//   agg   = segment_sum(msg, dst)                           [N=10000, 128]
//   out   = concat(x, agg) @ W_upd + b_upd                  [N, 128]
//
// Exact factorization (linearity of the message Linear):
//   msg(e)  = Ms[src[e]] + Md[dst[e]] + b_msg,
//     Ms = X @ W_msg[0:128,:]   (small WMMA GEMM, f32)
//     Md = X @ W_msg[128:256,:] (small WMMA GEMM, f32)
//   agg[n]  = scatter_add(Ms[src], dst)[n] + deg[n]*(Md[n] + b_msg)
// This removes the 42-GFLOP per-edge GEMM; the edge phase becomes an
// L2-resident gather (Ms, 5 MB) + 82M global_atomic_add_f32 into agg (5 MB),
// both far inside the 192 MB L2. The remaining GEMMs (~1.3 GFLOP) run on
// V_WMMA_F32_16X16X4_F32 so the f32 reference precision is preserved.
//
// Edge phase: 32 edges batched per wave; per-lane coalesced index loads,
// indices broadcast to SGPRs via v_readlane -> SADDR-form global gather +
// 4 non-returning global_atomic_add_f32 per lane per edge.
// ============================================================================

typedef __attribute__((ext_vector_type(2))) float v2f;
typedef __attribute__((ext_vector_type(8))) float v8f;

#define F 128  // IN_F == OUT_F == 128

// ---------------------------------------------------------------------------
// Zero agg[N*128] (as float4) and deg[N].
// ---------------------------------------------------------------------------
__global__ void zero_ws_kernel(float4* __restrict__ agg4, int* __restrict__ deg,
                               long long nAgg4, long long nDeg) {
    long long i = (long long)blockIdx.x * blockDim.x + threadIdx.x;
    long long stride = (long long)gridDim.x * blockDim.x;
    for (long long j = i; j < nAgg4; j += stride) agg4[j] = make_float4(0.f, 0.f, 0.f, 0.f);
    for (long long j = i; j < nDeg; j += stride) deg[j] = 0;
}

// ---------------------------------------------------------------------------
// One f32 WMMA step: A-frag (16x4) per ISA 7.12.2: lane L (L<16) holds
// A[m][k0..k0+1], lane L+16 holds A[m][k0+2..k0+3]; B-frag mirrored for 4x16.
// ---------------------------------------------------------------------------
__device__ __forceinline__ void wmma_step(const float* __restrict__ arow,
                                          const float* __restrict__ wcol,
                                          int k0, v8f& c) {
    v2f a = *(const v2f*)(arow + k0);                 // 8B-aligned (k0 even)
    v2f b;
    b.x = wcol[(size_t)k0 * F];
    b.y = wcol[(size_t)(k0 + 1) * F];
    c = __builtin_amdgcn_wmma_f32_16x16x4_f32(false, a, false, b,
                                              (short)0, c, false, false);
}

// ---------------------------------------------------------------------------
// C = A[N,128] @ W[128,128]  (row-major, W row stride 128).
// One wave per 16x16 output tile. grid = (128/16, ceil(N/16)), block = 32.
// ---------------------------------------------------------------------------
__global__ void __launch_bounds__(32)
wmma_gemm_f32(const float* __restrict__ A, const float* __restrict__ W,
              float* __restrict__ OUT, int N) {
    const int lane    = threadIdx.x;         // 0..31, EXEC all-ones for WMMA
    const int colBase = blockIdx.x << 4;
    const int rowBase = blockIdx.y << 4;
    const int mn      = lane & 15;           // A row / B col within tile
    const int kSub    = (lane >> 4) << 1;    // 0 (lanes 0-15) or 2 (16-31)

    int row = rowBase + mn;
    if (row >= N) row = N - 1;               // clamp loads; stores guarded
    const float* __restrict__ arow = A + (size_t)row * F;
    const float* __restrict__ wcol = W + colBase + mn;

    v8f c0 = {}, c1 = {};                    // 2 accumulators break C-chain
#pragma unroll
    for (int kb = 0; kb < F; kb += 8) {
        wmma_step(arow, wcol, kb + kSub,     c0);
        wmma_step(arow, wcol, kb + 4 + kSub, c1);
    }
    v8f c = c0 + c1;

    // D layout: VGPR r -> M = r + 8*(lane>=16), N = lane&15
    const int rOff = (lane >> 4) << 3;
#pragma unroll
    for (int r = 0; r < 8; ++r) {
        int orow = rowBase + rOff + r;
        if (orow < N) OUT[(size_t)orow * F + colBase + mn] = c[r];
    }
}

// ---------------------------------------------------------------------------
// Edge scatter, 32 edges per wave batch:
//   - lane e loads (src,dst) of edge base+e  (coalesced) and does one
//     deg[dst]++ atomic,
//   - the wave then walks the 32 edges; indices are broadcast into SGPRs via
//     v_readlane so gather/scatter use SADDR-form addressing,
//   - per edge: lane handles 4 consecutive floats: float4 gather from Ms[src]
//     + 4 non-returning global_atomic_add_f32 into agg[dst].
// Ms and agg are each 5 MB -> the whole phase lives in L2.
// ---------------------------------------------------------------------------
__global__ void edge_scatter_kernel(const int* __restrict__ ei,  // [2, E]
                                    const float* __restrict__ Ms,
                                    float* __restrict__ agg,
                                    int* __restrict__ deg, int E) {
    const int lane   = threadIdx.x & 31;
    const int wave   = blockIdx.x * (blockDim.x >> 5) + (threadIdx.x >> 5);
    const int nWaves = gridDim.x * (blockDim.x >> 5);
    const int lOff   = lane << 2;            // 4 floats per lane

    for (int base = wave * 32; base < E; base += nWaves * 32) {
        const int myE   = base + lane;
        const bool valid = myE < E;
        const int s = valid ? ei[myE]     : 0;
        const int d = valid ? ei[E + myE] : 0;
        if (valid) atomicAdd(deg + d, 1);

        const int cnt = (E - base < 32) ? (E - base) : 32;
        for (int j = 0; j < cnt; ++j) {
            const int sj = __builtin_amdgcn_readlane(s, j);  // -> SGPR
            const int dj = __builtin_amdgcn_readlane(d, j);  // -> SGPR
            const float4 v = *(const float4*)(Ms + (size_t)sj * F + lOff);
            float* dp = agg + (size_t)dj * F + lOff;
            atomicAdd(dp + 0, v.x);
            atomicAdd(dp + 1, v.y);
            atomicAdd(dp + 2, v.z);
            atomicAdd(dp + 3, v.w);
        }
    }
}

// ---------------------------------------------------------------------------
// Fused finalize + update GEMM:
//   out = concat(x, agg + deg*(Md + b_msg)) @ W_upd + b_upd
// K = 256 (first 128 from atom features, second 128 finalized on the fly).
// One wave per 16x16 output tile.
// ---------------------------------------------------------------------------
__global__ void __launch_bounds__(32)
wmma_update_f32(const float* __restrict__ atoms, const float* __restrict__ agg,
                const float* __restrict__ Md, const int* __restrict__ deg,
                const float* __restrict__ b_msg,
                const float* __restrict__ W_upd,   // [256, 128] row-major
                const float* __restrict__ b_upd,
                float* __restrict__ OUT, int N) {
    const int lane    = threadIdx.x;
    const int colBase = blockIdx.x << 4;
    const int rowBase = blockIdx.y << 4;
    const int mn      = lane & 15;
    const int kSub    = (lane >> 4) << 1;

    int row = rowBase + mn;
    if (row >= N) row = N - 1;
    const float* __restrict__ arow = atoms + (size_t)row * F;
    const float* __restrict__ grow = agg   + (size_t)row * F;
    const float* __restrict__ mrow = Md    + (size_t)row * F;
    const float  dF = (float)deg[row];
    const float* __restrict__ wcol = W_upd + colBase + mn;

    v8f c0 = {}, c1 = {};

    // K = 0..127 : atom features
#pragma unroll
    for (int kb = 0; kb < F; kb += 8) {
        wmma_step(arow, wcol, kb + kSub,     c0);
        wmma_step(arow, wcol, kb + 4 + kSub, c1);
    }
    // K = 128..255 : finalized aggregated messages
#pragma unroll
    for (int kb = 0; kb < F; kb += 4) {
        const int k0 = kb + kSub;
        v2f g  = *(const v2f*)(grow + k0);
        v2f m  = *(const v2f*)(mrow + k0);
        v2f bm = *(const v2f*)(b_msg + k0);
        v2f a;
        a.x = g.x + dF * (m.x + bm.x);
        a.y = g.y + dF * (m.y + bm.y);
        v2f b;
        b.x = wcol[(size_t)(F + k0) * F];
        b.y = wcol[(size_t)(F + k0 + 1) * F];
        v8f& c = (kb & 4) ? c1 : c0;
        c = __builtin_amdgcn_wmma_f32_16x16x4_f32(false, a, false, b,
                                                  (short)0, c, false, false);
    }
    v8f c = c0 + c1;

    const float bu   = b_upd[colBase + mn];
    const int   rOff = (lane >> 4) << 3;
#pragma unroll
    for (int r = 0; r < 8; ++r) {
        int orow = rowBase + rOff + r;
        if (orow < N) OUT[(size_t)orow * F + colBase + mn] = c[r] + bu;
    }
}

// ---------------------------------------------------------------------------
// Launch. Inputs (setup_inputs order):
//   0: atom_features [N,128] f32   1: edge_indices [2,E] int
//   2: W_msg [256,128] f32         3: b_msg [128] f32
//   4: W_upd [256,128] f32         5: b_upd [128] f32
// Workspace: Ms[N*128] | Md[N*128] | agg[N*128] (f32) | deg[N] (int) ~15.4 MB
// ---------------------------------------------------------------------------
extern "C" void kernel_launch(void* const* d_in, const int* in_sizes, int n_in,
                              void* d_out, int out_size, void* d_ws, size_t ws_size,
                              hipStream_t stream) {
    const float* atoms = (const float*)d_in[0];
    const int*   ei    = (const int*)d_in[1];
    const float* W_msg = (const float*)d_in[2];
    const float* b_msg = (const float*)d_in[3];
    const float* W_upd = (const float*)d_in[4];
    const float* b_upd = (const float*)d_in[5];
    float*       out   = (float*)d_out;

    const int N = in_sizes[0] / F;   // 10000
    const int E = in_sizes[1] / 2;   // 640000

    float* Ms  = (float*)d_ws;
    float* Md  = Ms + (size_t)N * F;
    float* agg = Md + (size_t)N * F;
    int*   deg = (int*)(agg + (size_t)N * F);

    // 1) zero agg + deg (workspace is poisoned by the harness)
    zero_ws_kernel<<<512, 256, 0, stream>>>((float4*)agg, deg,
                                            (long long)N * (F / 4), (long long)N);

    // 2) Ms = X @ W_msg[0:128,:], Md = X @ W_msg[128:256,:]
    dim3 tiles(F / 16, (N + 15) / 16);
    wmma_gemm_f32<<<tiles, 32, 0, stream>>>(atoms, W_msg,         Ms, N);
    wmma_gemm_f32<<<tiles, 32, 0, stream>>>(atoms, W_msg + F * F, Md, N);

    // 3) agg[dst] += Ms[src]; deg[dst]++   (L2-resident atomics, 32-edge batches)
    edge_scatter_kernel<<<2048, 256, 0, stream>>>(ei, Ms, agg, deg, E);

    // 4) out = concat(x, agg + deg*(Md+b_msg)) @ W_upd + b_upd
    wmma_update_f32<<<tiles, 32, 0, stream>>>(atoms, agg, Md, deg, b_msg,
                                              W_upd, b_upd, out, N);
}